// Net_46849503265421
// MI455X (gfx1250) — compile-verified
//
#include <hip/hip_runtime.h>

// ---------------------------------------------------------------------------
// GCN (3x gcn_conv, H=32) for MI455X / gfx1250.
//  - Linear transforms use V_WMMA_F32_16X16X4_F32 (exact fp32 path).
//  - Edge scatter: one wave32 per edge, one float per lane -> coalesced
//    128B gather + 32 global_atomic_add_f32 (L2-resident working set).
// ---------------------------------------------------------------------------

typedef float v2f __attribute__((ext_vector_type(2)));
typedef float v8f __attribute__((ext_vector_type(8)));

#define HID 32

// ---------------- degree / norm pipeline ----------------

__global__ void k_init_deg(float* __restrict__ deg, int n) {
    int i = blockIdx.x * blockDim.x + threadIdx.x;
    if (i < n) deg[i] = 1.0f;  // self-loop
}

__global__ void k_accum_deg(const int* __restrict__ dst, float* __restrict__ deg, int e) {
    int i = blockIdx.x * blockDim.x + threadIdx.x;
    if (i < e) atomicAdd(&deg[dst[i]], 1.0f);
}

__global__ void k_dinv(const float* __restrict__ deg, float* __restrict__ dinv, int n) {
    int i = blockIdx.x * blockDim.x + threadIdx.x;
    if (i < n) dinv[i] = rsqrtf(deg[i]);
}

__global__ void k_edge_norm(const int* __restrict__ src, const int* __restrict__ dst,
                            const float* __restrict__ dinv, float* __restrict__ nrm, int e) {
    int i = blockIdx.x * blockDim.x + threadIdx.x;
    if (i < e) nrm[i] = dinv[src[i]] * dinv[dst[i]];
}

// ---------------- fc1: [N,2] @ [2,32] + b, ReLU ----------------

__global__ void k_fc1(const float* __restrict__ x, const float* __restrict__ t,
                      const float* __restrict__ w, const float* __restrict__ b,
                      float* __restrict__ X, int n) {
    int i = blockIdx.x * blockDim.x + threadIdx.x;  // node*32 + h
    if (i < n * HID) {
        int node = i >> 5, h = i & 31;
        float v = x[node] * w[h] + t[node] * w[HID + h] + b[h];
        X[i] = fmaxf(v, 0.0f);
    }
}

// ---------------- WMMA GEMM: H = relu(Xin) @ W   [N,32]x[32,32] ----------------
// One wave computes a 16x32 output tile: 2 N-tiles x 8 K-chunks of
// v_wmma_f32_16x16x4_f32. ReLU folded into A load (idempotent for layer 1).

__global__ void k_gemm_wmma(const float* __restrict__ Xin, const float* __restrict__ W,
                            float* __restrict__ Hout, int n_nodes) {
    __shared__ float lW[HID * HID];  // 4 KB: stage weights once per block
    for (int i = threadIdx.x; i < HID * HID; i += blockDim.x) lW[i] = W[i];
    __syncthreads();

    int wave = (int)((blockIdx.x * blockDim.x + threadIdx.x) >> 5);
    int lane = threadIdx.x & 31;
    int row0 = wave * 16;
    if (row0 >= n_nodes) return;  // wave-uniform: EXEC stays all-ones for WMMA

    int lrow = lane & 15;   // row (A) / column (B,C,D) within tile
    int half = lane >> 4;   // lane-half selects K pair / row group

    // B fragments: VGPR0 holds K=k0+0 (lanes 0-15) / K=k0+2 (lanes 16-31), VGPR1 K=+1/+3
    v2f bfrag[2][8];
#pragma unroll
    for (int nt = 0; nt < 2; ++nt)
#pragma unroll
        for (int kc = 0; kc < 8; ++kc) {
            int col = nt * 16 + lrow;
            int k0  = kc * 4 + 2 * half;
            bfrag[nt][kc].x = lW[(k0 + 0) * HID + col];
            bfrag[nt][kc].y = lW[(k0 + 1) * HID + col];
        }

    const float* rowp = Xin + (size_t)(row0 + lrow) * HID;
    v8f c0 = {};
    v8f c1 = {};
#pragma unroll
    for (int kc = 0; kc < 8; ++kc) {
        int k0 = kc * 4 + 2 * half;
        v2f a;
        a.x = fmaxf(rowp[k0 + 0], 0.0f);
        a.y = fmaxf(rowp[k0 + 1], 0.0f);
        c0 = __builtin_amdgcn_wmma_f32_16x16x4_f32(false, a, false, bfrag[0][kc],
                                                   (short)0, c0, false, false);
        c1 = __builtin_amdgcn_wmma_f32_16x16x4_f32(false, a, false, bfrag[1][kc],
                                                   (short)0, c1, false, false);
    }

    // D layout: VGPR i holds rows (i, i+8) split across lane halves, N = lrow
#pragma unroll
    for (int i = 0; i < 8; ++i) {
        size_t r = (size_t)(row0 + i + 8 * half) * HID;
        Hout[r + 0 * 16 + lrow] = c0[i];
        Hout[r + 1 * 16 + lrow] = c1[i];
    }
}

// ---------------- seed: agg = h * (1/deg) + bias ----------------

__global__ void k_seed(const float* __restrict__ h, const float* __restrict__ deg,
                       const float* __restrict__ bias, float* __restrict__ agg, int n) {
    int i = blockIdx.x * blockDim.x + threadIdx.x;
    if (i < n * HID) {
        int node = i >> 5, j = i & 31;
        agg[i] = h[i] * (1.0f / deg[node]) + bias[j];
    }
}

// ---------------- edge scatter: one wave per edge ----------------

__global__ void k_scatter(const int* __restrict__ src, const int* __restrict__ dst,
                          const float* __restrict__ nrm, const float* __restrict__ h,
                          float* __restrict__ agg, int e) {
    int gid  = blockIdx.x * blockDim.x + threadIdx.x;
    int edge = gid >> 5;
    int lane = gid & 31;
    if (edge < e) {
        int s   = src[edge];
        int d   = dst[edge];
        float w = nrm[edge];
        float v = h[(size_t)s * HID + lane] * w;        // coalesced 128B gather
        atomicAdd(&agg[(size_t)d * HID + lane], v);     // coalesced 128B scatter-add
    }
}

// ---------------- fc3: out = relu(agg) @ [32,1] + b ----------------

__global__ void k_fc3(const float* __restrict__ agg, const float* __restrict__ w,
                      const float* __restrict__ b, float* __restrict__ out, int n) {
    int i = blockIdx.x * blockDim.x + threadIdx.x;
    if (i < n) {
        float acc = b[0];
        const float* row = agg + (size_t)i * HID;
#pragma unroll
        for (int j = 0; j < HID; ++j) acc += fmaxf(row[j], 0.0f) * w[j];
        out[i] = acc;
    }
}

// ---------------------------------------------------------------------------

static inline int cdiv(long long a, long long b) { return (int)((a + b - 1) / b); }

extern "C" void kernel_launch(void* const* d_in, const int* in_sizes, int n_in,
                              void* d_out, int out_size, void* d_ws, size_t ws_size,
                              hipStream_t stream) {
    const float* x     = (const float*)d_in[0];
    const float* t     = (const float*)d_in[1];
    const int*   ei    = (const int*)d_in[2];   // [2,E] row-major
    const float* fc1_w = (const float*)d_in[3];
    const float* fc1_b = (const float*)d_in[4];
    const float* w1    = (const float*)d_in[5];
    const float* b1    = (const float*)d_in[6];
    const float* w2    = (const float*)d_in[7];
    const float* b2    = (const float*)d_in[8];
    const float* w3    = (const float*)d_in[9];
    const float* b3    = (const float*)d_in[10];
    const float* fc3_w = (const float*)d_in[11];
    const float* fc3_b = (const float*)d_in[12];
    float* out = (float*)d_out;

    const int N = in_sizes[0];           // 100000
    const int E = in_sizes[2] / 2;       // 2500000
    const int* src = ei;
    const int* dst = ei + E;

    // workspace layout (floats): A, B, H (N*32 each), deg, dinv (N), norm (E)
    float* bufA = (float*)d_ws;
    float* bufB = bufA + (size_t)N * HID;
    float* bufH = bufB + (size_t)N * HID;
    float* deg  = bufH + (size_t)N * HID;
    float* dinv = deg + N;
    float* nrm  = dinv + N;

    const int TB = 256;
    const int nBlkN   = cdiv(N, TB);
    const int nBlkE   = cdiv(E, TB);
    const int nBlkNH  = cdiv((long long)N * HID, TB);
    const int nBlkEW  = cdiv((long long)E * 32, TB);              // wave per edge
    const int nBlkGemm = cdiv((long long)cdiv(N, 16) * 32, TB);   // wave per 16 rows

    // degrees + normalization
    k_init_deg<<<nBlkN, TB, 0, stream>>>(deg, N);
    k_accum_deg<<<nBlkE, TB, 0, stream>>>(dst, deg, E);
    k_dinv<<<nBlkN, TB, 0, stream>>>(deg, dinv, N);
    k_edge_norm<<<nBlkE, TB, 0, stream>>>(src, dst, dinv, nrm, E);

    // fc1 + relu -> A
    k_fc1<<<nBlkNH, TB, 0, stream>>>(x, t, fc1_w, fc1_b, bufA, N);

    // layer 1: A -> B
    k_gemm_wmma<<<nBlkGemm, TB, 0, stream>>>(bufA, w1, bufH, N);
    k_seed<<<nBlkNH, TB, 0, stream>>>(bufH, deg, b1, bufB, N);
    k_scatter<<<nBlkEW, TB, 0, stream>>>(src, dst, nrm, bufH, bufB, E);

    // layer 2: B -> A (relu folded into gemm A-load)
    k_gemm_wmma<<<nBlkGemm, TB, 0, stream>>>(bufB, w2, bufH, N);
    k_seed<<<nBlkNH, TB, 0, stream>>>(bufH, deg, b2, bufA, N);
    k_scatter<<<nBlkEW, TB, 0, stream>>>(src, dst, nrm, bufH, bufA, E);

    // layer 3: A -> B
    k_gemm_wmma<<<nBlkGemm, TB, 0, stream>>>(bufA, w3, bufH, N);
    k_seed<<<nBlkNH, TB, 0, stream>>>(bufH, deg, b3, bufB, N);
    k_scatter<<<nBlkEW, TB, 0, stream>>>(src, dst, nrm, bufH, bufB, E);

    // fc3 (relu folded in) -> out
    k_fc3<<<nBlkN, TB, 0, stream>>>(bufB, fc3_w, fc3_b, out, N);

    (void)n_in; (void)out_size; (void)ws_size; (void)fc3_b;
}